// LogicTreeConv2d_12111807775266
// MI455X (gfx1250) — compile-verified
//
#include <hip/hip_runtime.h>
#include <math.h>
#include <stdint.h>

// ---------------- problem constants ----------------
#define OUT_CH     256
#define IN_CH      64
#define NUM_NODES  7
#define NUM_LEAVES 8
#define IMG_H      48
#define IMG_W      48
#define BATCH      16

// ---------------- tiling ----------------
#define TH        2                    // output rows per block
#define ROWSP     (TH + 2)             // staged rows incl. halo
#define XP        56                   // padded LDS row stride (floats); data at [4..51]
#define NROWBLK   (IMG_H / TH)         // 24 row-blocks
#define TILE_FLOATS (IN_CH * ROWSP * XP)   // 14336 floats = 56 KB
#define BLK       384                  // 12 wave32
#define PIX       (TH * IMG_W)         // 96 pixels per tile
#define NOGRP     (BLK / PIX)          // 4 channel groups
#define OPER      (OUT_CH / NOGRP)     // 64 channels per thread
#define NROWS_W   (OUT_CH * NUM_NODES) // 1792 weight rows

// ---------------- feature detection ----------------
#if defined(__has_builtin)
#  if __has_builtin(__builtin_amdgcn_global_load_async_to_lds_b128)
#    define HAVE_ASYNC_LDS 1
#  endif
#  if __has_builtin(__builtin_amdgcn_wmma_f32_16x16x4_f32)
#    define HAVE_WMMA_F32K4 1
#  endif
#endif
#ifndef HAVE_ASYNC_LDS
#  define HAVE_ASYNC_LDS 0
#endif
#ifndef HAVE_WMMA_F32K4
#  define HAVE_WMMA_F32K4 0
#endif

typedef __attribute__((ext_vector_type(2))) float v2f;
typedef __attribute__((ext_vector_type(8))) float v8f;
// async-LDS builtin takes pointers to 16-byte int vectors in AS1 / AS3
typedef int v4i __attribute__((vector_size(16)));
typedef __attribute__((address_space(1))) v4i* g1p_t;
typedef __attribute__((address_space(3))) v4i* l3p_t;

__constant__ float GATE_COEFF[16][4] = {
    {0.f, 0.f, 0.f, 0.f}, {0.f, 0.f, 0.f, 1.f}, {0.f, 1.f, 0.f, -1.f}, {0.f, 1.f, 0.f, 0.f},
    {0.f, 0.f, 1.f, -1.f}, {0.f, 0.f, 1.f, 0.f}, {0.f, 1.f, 1.f, -2.f}, {0.f, 1.f, 1.f, -1.f},
    {1.f, -1.f, -1.f, 1.f}, {1.f, -1.f, -1.f, 2.f}, {1.f, 0.f, -1.f, 0.f}, {1.f, 0.f, -1.f, 1.f},
    {1.f, -1.f, 0.f, 0.f}, {1.f, -1.f, 0.f, 1.f}, {1.f, 0.f, 0.f, -1.f}, {1.f, 0.f, 0.f, 0.f}};

// ================================================================
// Kernel 1: gatew[row][0..3] = softmax(logits[row][:16]) @ GATE_COEFF
// One wave32 per 16 rows; K=16 split into 4 chunks of V_WMMA_F32_16X16X4_F32.
// ================================================================
__global__ __launch_bounds__(32) void gate_weight_kernel(
    const float* __restrict__ logits, float* __restrict__ gatew) {
  __shared__ float sm[16][17];   // softmax probs, padded stride (bank-conflict free)
  const int lane    = threadIdx.x;
  const int rowBase = blockIdx.x * 16;

  if (lane < 16) {
    const float* lp = logits + (size_t)(rowBase + lane) * 16;
    float v[16];
    float mx = -3.402823466e38f;
#pragma unroll
    for (int g = 0; g < 16; ++g) { v[g] = lp[g]; mx = fmaxf(mx, v[g]); }
    float s = 0.f;
#pragma unroll
    for (int g = 0; g < 16; ++g) { v[g] = expf(v[g] - mx); s += v[g]; }
    const float inv = 1.f / s;
#pragma unroll
    for (int g = 0; g < 16; ++g) sm[lane][g] = v[g] * inv;
  }
  __syncthreads();

#if HAVE_WMMA_F32K4
  // A: 16x4 f32 fragment; lanes 0-15 hold K={0,1}, lanes 16-31 hold K={2,3}, M=lane&15.
  // B: 4x16 f32 fragment; lanes 0-15 hold rows K={0,1}, lanes 16-31 rows K={2,3}, N=lane&15.
  const int M  = lane & 15;
  const int hi = lane >> 4;      // 0 or 1
  v8f acc = {};
#pragma unroll
  for (int q = 0; q < 4; ++q) {
    v2f a, b;
    a.x = sm[M][4 * q + 2 * hi + 0];
    a.y = sm[M][4 * q + 2 * hi + 1];
    b.x = (M < 4) ? GATE_COEFF[4 * q + 2 * hi + 0][M] : 0.f;
    b.y = (M < 4) ? GATE_COEFF[4 * q + 2 * hi + 1][M] : 0.f;
    acc = __builtin_amdgcn_wmma_f32_16x16x4_f32(false, a, false, b,
                                                (short)0, acc, false, false);
  }
  // D layout: vgpr v -> row (v + 8*hi), col = lane&15. Keep cols 0..3.
#pragma unroll
  for (int v = 0; v < 8; ++v) {
    const int row = v + 8 * hi;
    if (M < 4) gatew[(size_t)(rowBase + row) * 4 + M] = acc[v];
  }
#else
  if (lane < 16) {
#pragma unroll
    for (int n = 0; n < 4; ++n) {
      float s = 0.f;
#pragma unroll
      for (int g = 0; g < 16; ++g) s += sm[lane][g] * GATE_COEFF[g][n];
      gatew[(size_t)(rowBase + lane) * 4 + n] = s;
    }
  }
#endif
}

// ================================================================
// Kernel 2: decode leaf indices -> LDS float offsets
// li -> (c = li/9, i = (li%9)/3, j = li%3); off = (c*ROWSP + i)*XP + j + 3
// ================================================================
__global__ void leaf_off_kernel(const int* __restrict__ lidx,
                                int* __restrict__ loff, int n) {
  const int e = blockIdx.x * blockDim.x + threadIdx.x;
  if (e >= n) return;
  const int l = lidx[e];
  const int c = l / 9;
  const int k = l - 9 * c;
  const int i = k / 3;
  const int j = k - 3 * i;
  loff[e] = (c * ROWSP + i) * XP + j + 3;
}

// ================================================================
// Kernel 3: main logic-tree conv
// ================================================================
__device__ __forceinline__ float gate4(const float* __restrict__ w, float a, float b) {
  return fmaf(w[3], a * b, fmaf(w[2], b, fmaf(w[1], a, w[0])));
}

__global__ __launch_bounds__(BLK) void logictree_kernel(
    const float* __restrict__ x, const float* __restrict__ gatew,
    const int* __restrict__ loff, float* __restrict__ out) {
  __shared__ __align__(16) float tile[TILE_FLOATS];   // 56 KB

  const int t  = threadIdx.x;
  const int b  = blockIdx.x / NROWBLK;
  const int y0 = (blockIdx.x % NROWBLK) * TH;

  // ---- zero LDS tile (halo columns/rows stay zero) ----
  {
    float4* tz = (float4*)tile;
    const int n4 = TILE_FLOATS / 4;        // 3584
    for (int i = t; i < n4; i += BLK) tz[i] = make_float4(0.f, 0.f, 0.f, 0.f);
  }
  __syncthreads();

  // ---- stage valid region: 64 ch x up-to-4 rows x 48 cols (12 float4 / row) ----
  const int NF4 = IN_CH * ROWSP * (IMG_W / 4);   // 3072 b128 transfers
  for (int e = t; e < NF4; e += BLK) {
    const int c   = e / (ROWSP * 12);
    const int rem = e - c * (ROWSP * 12);
    const int yy  = rem / 12;
    const int xi  = rem - yy * 12;
    const int gy  = y0 - 1 + yy;
    if ((unsigned)gy < (unsigned)IMG_H) {
      const float* src = x + (((size_t)(b * IN_CH + c) * IMG_H + gy) * IMG_W + 4 * xi);
      float* dst = &tile[(c * ROWSP + yy) * XP + 4 + 4 * xi];
#if HAVE_ASYNC_LDS
      __builtin_amdgcn_global_load_async_to_lds_b128(
          (g1p_t)(uintptr_t)src,
          (l3p_t)(uint32_t)(uintptr_t)dst, 0, 0);
#else
      *(float4*)dst = *(const float4*)src;
#endif
    }
  }
#if HAVE_ASYNC_LDS
#  if defined(__has_builtin) && __has_builtin(__builtin_amdgcn_s_wait_asynccnt)
  __builtin_amdgcn_s_wait_asynccnt(0);
#  else
  asm volatile("s_wait_asynccnt 0" ::: "memory");
#  endif
#endif
  __syncthreads();

  // ---- compute: thread owns one pixel and OPER output channels ----
  const int p     = t % PIX;                 // pixel in tile
  const int ogrp  = __builtin_amdgcn_readfirstlane(t / PIX);  // wave-uniform
  const int py    = p / IMG_W;
  const int px    = p - py * IMG_W;
  const int pbase = py * XP + px;
  const int oBeg  = ogrp * OPER;

  float* outp = out + (size_t)b * OUT_CH * IMG_H * IMG_W +
                (size_t)(y0 + py) * IMG_W + px;

  for (int o = oBeg; o < oBeg + OPER; ++o) {
    const int*   lo = loff + o * NUM_LEAVES;
    const float* wv = gatew + (size_t)o * NUM_NODES * 4;

    const float v0 = tile[pbase + lo[0]];
    const float v1 = tile[pbase + lo[1]];
    const float v2 = tile[pbase + lo[2]];
    const float v3 = tile[pbase + lo[3]];
    const float v4 = tile[pbase + lo[4]];
    const float v5 = tile[pbase + lo[5]];
    const float v6 = tile[pbase + lo[6]];
    const float v7 = tile[pbase + lo[7]];

    const float n0 = gate4(wv + 0,  v0, v1);
    const float n1 = gate4(wv + 4,  v2, v3);
    const float n2 = gate4(wv + 8,  v4, v5);
    const float n3 = gate4(wv + 12, v6, v7);
    const float m0 = gate4(wv + 16, n0, n1);
    const float m1 = gate4(wv + 20, n2, n3);
    const float r  = gate4(wv + 24, m0, m1);

    outp[(size_t)o * IMG_H * IMG_W] = r;
  }
}

// ================================================================
// Host launcher
// ================================================================
extern "C" void kernel_launch(void* const* d_in, const int* in_sizes, int n_in,
                              void* d_out, int out_size, void* d_ws, size_t ws_size,
                              hipStream_t stream) {
  (void)in_sizes; (void)n_in; (void)out_size; (void)ws_size;

  const float* x      = (const float*)d_in[0];   // (16, 64, 48, 48) f32
  const float* logits = (const float*)d_in[1];   // (256, 7, 16) f32
  const int*   leaf   = (const int*)d_in[2];     // (256, 8) i32
  float*       out    = (float*)d_out;           // (16, 256, 48, 48) f32

  float* gatew = (float*)d_ws;                              // 1792*4 f32 = 28672 B
  int*   loff  = (int*)((char*)d_ws + NROWS_W * 4 * sizeof(float));  // 2048 i32

  gate_weight_kernel<<<NROWS_W / 16, 32, 0, stream>>>(logits, gatew);
  leaf_off_kernel<<<(OUT_CH * NUM_LEAVES + 255) / 256, 256, 0, stream>>>(
      leaf, loff, OUT_CH * NUM_LEAVES);
  logictree_kernel<<<BATCH * NROWBLK, BLK, 0, stream>>>(x, gatew, loff, out);
}